// Loss_34909494182278
// MI455X (gfx1250) — compile-verified
//
#include <hip/hip_runtime.h>
#include <stdint.h>
#include <math.h>

// Problem geometry (matches reference): B rows of S scores each.
#define SLEN     8192
#define THREADS  256
#define WAVES    (THREADS / 32)
#define PER_THR  (SLEN / THREADS)   // 32 elements per thread, held in registers

// Order-preserving float -> uint key (ascending float => ascending uint)
__device__ __forceinline__ unsigned f2key(float f) {
    unsigned u = __float_as_uint(f);
    return (u & 0x80000000u) ? ~u : (u ^ 0x80000000u);
}
__device__ __forceinline__ float key2f(unsigned k) {
    unsigned u = (k & 0x80000000u) ? (k ^ 0x80000000u) : ~k;
    return __uint_as_float(u);
}

__global__ __launch_bounds__(THREADS)
void topk_bce_row_kernel(const float* __restrict__ scores,
                         const float* __restrict__ label,
                         const int*   __restrict__ seqlen,
                         float*       __restrict__ row_loss) {
    __shared__ float s_row[SLEN];          // 32 KB staged row
    __shared__ int   s_cnt[WAVES];
    __shared__ float s_fsum[WAVES];
    __shared__ int   s_cgt[WAVES];

    const int b    = blockIdx.x;
    const int tid  = threadIdx.x;
    const int lane = tid & 31;
    const int wave = tid >> 5;

    const float* src = scores + (size_t)b * SLEN;

    // ---- CDNA5 async DMA: global -> LDS, b128 per lane per issue ----
    // Each thread issues 8 x 16B; whole row covered by the block.
    #pragma unroll
    for (int i = 0; i < PER_THR / 4; ++i) {
        const int elem = (i * THREADS + tid) * 4;
        // Low 32 bits of a flat LDS pointer are the wave-relative LDS byte address.
        unsigned lds_off = (unsigned)(size_t)(&s_row[elem]);
        const float* g = src + elem;
        asm volatile("global_load_async_to_lds_b128 %0, %1, off"
                     :: "v"(lds_off), "v"(g) : "memory");
    }
    asm volatile("s_wait_asynccnt 0" ::: "memory");
    __syncthreads();

    const int L = seqlen[b];
    const unsigned k = (unsigned)(L / 16 + 1);   // reference: seqlen//16 + 1

    // ---- Pull keys into registers (wave w owns contiguous 1024-elem chunk) ----
    unsigned keys[PER_THR];
    {
        const int base = wave * (PER_THR * 32) + lane;
        #pragma unroll
        for (int j = 0; j < PER_THR; ++j) {
            const int idx = base + j * 32;       // stride-32: conflict-free LDS
            const float f = s_row[idx];
            keys[j] = (idx < L) ? f2key(f) : 0u; // padding below every real key
        }
    }

    // ---- Bitwise radix select of the k-th largest key ----
    unsigned prefix = 0;
    unsigned remaining = k;
    for (int bit = 31; bit >= 0; --bit) {
        const unsigned want = (prefix | (1u << bit)) >> bit;
        int c = 0;
        #pragma unroll
        for (int j = 0; j < PER_THR; ++j) c += (int)((keys[j] >> bit) == want);
        #pragma unroll
        for (int off = 16; off >= 1; off >>= 1) c += __shfl_xor(c, off, 32);
        if (lane == 0) s_cnt[wave] = c;
        __syncthreads();
        int total = 0;
        #pragma unroll
        for (int w = 0; w < WAVES; ++w) total += s_cnt[w];
        __syncthreads();
        if ((unsigned)total >= remaining) prefix |= (1u << bit);
        else                              remaining -= (unsigned)total;
    }
    const unsigned thr_key = prefix;

    // ---- Sum of strictly-greater elements; ties filled with threshold ----
    float fsum = 0.0f;
    int   cgt  = 0;
    #pragma unroll
    for (int j = 0; j < PER_THR; ++j) {
        if (keys[j] > thr_key) { fsum += key2f(keys[j]); ++cgt; }
    }
    #pragma unroll
    for (int off = 16; off >= 1; off >>= 1) {
        fsum += __shfl_xor(fsum, off, 32);
        cgt  += __shfl_xor(cgt,  off, 32);
    }
    if (lane == 0) { s_fsum[wave] = fsum; s_cgt[wave] = cgt; }
    __syncthreads();

    if (tid == 0) {
        float Sg = 0.0f; int Cg = 0;
        #pragma unroll
        for (int w = 0; w < WAVES; ++w) { Sg += s_fsum[w]; Cg += s_cgt[w]; }
        const float thr = key2f(thr_key);
        const float topk_sum  = Sg + (float)((int)k - Cg) * thr;
        const float topk_mean = topk_sum / (float)k;
        const float inst = 1.0f / (1.0f + expf(-topk_mean));
        const float y = label[b];
        const float eps = 1e-12f;
        // Store the un-negated BCE term; final kernel applies -mean.
        row_loss[b] = y * logf(inst + eps) + (1.0f - y) * logf(1.0f - inst + eps);
    }
}

__global__ __launch_bounds__(256)
void bce_reduce_kernel(const float* __restrict__ row_loss,
                       float* __restrict__ out, int B) {
    __shared__ float s_p[8];
    const int tid  = threadIdx.x;
    const int lane = tid & 31;
    const int wave = tid >> 5;
    float acc = 0.0f;
    for (int i = tid; i < B; i += 256) acc += row_loss[i];   // fixed order
    #pragma unroll
    for (int off = 16; off >= 1; off >>= 1) acc += __shfl_xor(acc, off, 32);
    if (lane == 0) s_p[wave] = acc;
    __syncthreads();
    if (tid == 0) {
        float t = 0.0f;
        #pragma unroll
        for (int w = 0; w < 8; ++w) t += s_p[w];             // fixed order
        out[0] = -t / (float)B;
    }
}

extern "C" void kernel_launch(void* const* d_in, const int* in_sizes, int n_in,
                              void* d_out, int out_size, void* d_ws, size_t ws_size,
                              hipStream_t stream) {
    const float* scores = (const float*)d_in[0];  // [B, S, 1] f32
    const float* label  = (const float*)d_in[1];  // [B] f32
    const int*   seqlen = (const int*)d_in[2];    // [B] i32
    const int B = in_sizes[1];                    // label count == B

    float* row_loss = (float*)d_ws;               // B floats of scratch

    topk_bce_row_kernel<<<B, THREADS, 0, stream>>>(scores, label, seqlen, row_loss);
    bce_reduce_kernel<<<1, 256, 0, stream>>>(row_loss, (float*)d_out, B);
}